// WeightedProcrustes_55241869361483
// MI455X (gfx1250) — compile-verified
//
#include <hip/hip_runtime.h>

typedef float v2f __attribute__((ext_vector_type(2)));
typedef float v4f __attribute__((ext_vector_type(4)));
typedef float v8f __attribute__((ext_vector_type(8)));

#define WEIGHT_THRESH_F 0.5f
#define EPS_NORM 1e-5f

#if defined(__gfx1250__) && __has_builtin(__builtin_amdgcn_wmma_f32_16x16x4_f32)
#define HAVE_WMMA_F32X4 1
#else
#define HAVE_WMMA_F32X4 0
#endif

__device__ __forceinline__ v4f ntload4(const float* p) {
  return __builtin_nontemporal_load((const v4f*)p);
}

__device__ __forceinline__ float waveReduceSum(float x) {
#pragma unroll
  for (int off = 16; off > 0; off >>= 1)
    x += __shfl_down(x, off, 32);
  return x;
}

__global__ __launch_bounds__(256) void procrustes_fused(
    const float* __restrict__ src, const float* __restrict__ tgt,
    const float* __restrict__ wts, float* __restrict__ outR,
    float* __restrict__ outT, int N) {
  const int b = blockIdx.x;
  const int tid = threadIdx.x;
  const int lane = tid & 31;
  const int wave = tid >> 5;

  const float* __restrict__ sb = src + (size_t)b * N * 3;
  const float* __restrict__ tb = tgt + (size_t)b * N * 3;
  const float* __restrict__ wb = wts + (size_t)b * N;

  // 16 streaming accumulators: M'(9), P(3), Q(3), S(1)
  float m00 = 0.f, m01 = 0.f, m02 = 0.f;
  float m10 = 0.f, m11 = 0.f, m12 = 0.f;
  float m20 = 0.f, m21 = 0.f, m22 = 0.f;
  float p0 = 0.f, p1 = 0.f, p2 = 0.f;
  float q0 = 0.f, q1 = 0.f, q2 = 0.f;
  float sw = 0.f;

#define ACC(w, sx, sy, sz, tx, ty, tz)                                          \
  {                                                                             \
    float wsx = (w) * (sx), wsy = (w) * (sy), wsz = (w) * (sz);                 \
    m00 = fmaf(wsx, (tx), m00); m01 = fmaf(wsx, (ty), m01);                     \
    m02 = fmaf(wsx, (tz), m02);                                                 \
    m10 = fmaf(wsy, (tx), m10); m11 = fmaf(wsy, (ty), m11);                     \
    m12 = fmaf(wsy, (tz), m12);                                                 \
    m20 = fmaf(wsz, (tx), m20); m21 = fmaf(wsz, (ty), m21);                     \
    m22 = fmaf(wsz, (tz), m22);                                                 \
    p0 += wsx; p1 += wsy; p2 += wsz;                                            \
    q0 = fmaf((w), (tx), q0); q1 = fmaf((w), (ty), q1);                         \
    q2 = fmaf((w), (tz), q2);                                                   \
    sw += (w);                                                                  \
  }

  // Vector path: 4 points per thread-iteration, all traffic as b128 NT loads.
  const int nChunks = ((N & 3) == 0) ? (N >> 2) : 0;
  for (int c = tid; c < nChunks; c += 256) {
    const int base = c << 2;
    // stream-ahead hints (global_prefetch_b8); speculative, OOB is dropped
    __builtin_prefetch(sb + 3 * (base + 4096), 0, 0);
    __builtin_prefetch(tb + 3 * (base + 4096), 0, 0);
    __builtin_prefetch(wb + (base + 4096), 0, 0);

    v4f w4 = ntload4(wb + base);
    v4f sA = ntload4(sb + 3 * base);
    v4f sB = ntload4(sb + 3 * base + 4);
    v4f sC = ntload4(sb + 3 * base + 8);
    v4f tA = ntload4(tb + 3 * base);
    v4f tB = ntload4(tb + 3 * base + 4);
    v4f tC = ntload4(tb + 3 * base + 8);

    float w0 = (w4.x < WEIGHT_THRESH_F) ? 0.0f : w4.x;
    float w1 = (w4.y < WEIGHT_THRESH_F) ? 0.0f : w4.y;
    float w2 = (w4.z < WEIGHT_THRESH_F) ? 0.0f : w4.z;
    float w3 = (w4.w < WEIGHT_THRESH_F) ? 0.0f : w4.w;

    ACC(w0, sA.x, sA.y, sA.z, tA.x, tA.y, tA.z);
    ACC(w1, sA.w, sB.x, sB.y, tA.w, tB.x, tB.y);
    ACC(w2, sB.z, sB.w, sC.x, tB.z, tB.w, tC.x);
    ACC(w3, sC.y, sC.z, sC.w, tC.y, tC.z, tC.w);
  }
  // Scalar tail (covers everything if N % 4 != 0)
  for (int n = (nChunks << 2) + tid; n < N; n += 256) {
    float w = wb[n];
    w = (w < WEIGHT_THRESH_F) ? 0.0f : w;
    float sx = sb[3 * n + 0], sy = sb[3 * n + 1], sz = sb[3 * n + 2];
    float tx = tb[3 * n + 0], ty = tb[3 * n + 1], tz = tb[3 * n + 2];
    ACC(w, sx, sy, sz, tx, ty, tz);
  }
#undef ACC

  float acc[16] = {m00, m01, m02, m10, m11, m12, m20, m21, m22,
                   p0,  p1,  p2,  q0,  q1,  q2,  sw};
#pragma unroll
  for (int j = 0; j < 16; ++j) acc[j] = waveReduceSum(acc[j]);

  __shared__ float Wp[8 * 16];  // [wave][acc]
  __shared__ float tot[16];

  if (lane == 0) {
#pragma unroll
    for (int j = 0; j < 16; ++j) Wp[wave * 16 + j] = acc[j];
  }
  __syncthreads();

#if HAVE_WMMA_F32X4
  // Reduce the 8x16 wave-partials matrix with two chained V_WMMA_F32_16X16X4_F32.
  // A[m][k] = Wp[k][m] (ISA 16x4 f32 A layout: lane = m + 16*(k>=2), vgpr = k&1),
  // B = all-ones (layout-independent), so every column of D = row sums.
  if (wave == 0) {
    v8f c = {};
    v2f bOnes = {1.0f, 1.0f};
    const int m = lane & 15;
    const int kh = (lane >> 4) * 2;
#pragma unroll
    for (int it = 0; it < 2; ++it) {
      int k0 = it * 4 + kh;
      v2f a = {Wp[k0 * 16 + m], Wp[(k0 + 1) * 16 + m]};
      c = __builtin_amdgcn_wmma_f32_16x16x4_f32(false, a, false, bOnes,
                                                (short)0, c, false, false);
    }
    // C layout: lane L holds D[v + 8*(L/16)][L%16] in c[v]; read column 0.
    if (m == 0) {
      int rbase = (lane >> 4) * 8;
#pragma unroll
      for (int v = 0; v < 8; ++v) tot[rbase + v] = c[v];
    }
  }
#else
  if (tid < 16) {
    float s = 0.f;
#pragma unroll
    for (int k = 0; k < 8; ++k) s += Wp[k * 16 + tid];
    tot[tid] = s;
  }
#endif
  __syncthreads();

  // ---- per-batch finalize: 3x3 SVD (Jacobi on H^T H) + Kabsch ----
  if (tid == 0) {
    const float inv = 1.0f / (tot[15] + EPS_NORM);
    const float csum = tot[15] * inv;  // sum of normalized weights
    float mus[3] = {tot[9] * inv, tot[10] * inv, tot[11] * inv};
    float mut[3] = {tot[12] * inv, tot[13] * inv, tot[14] * inv};
    const float kc = 2.0f - csum;
    float Hm[3][3];
#pragma unroll
    for (int i = 0; i < 3; ++i)
#pragma unroll
      for (int j = 0; j < 3; ++j)
        Hm[i][j] = tot[i * 3 + j] * inv - kc * mus[i] * mut[j];

    float Amat[3][3];
#pragma unroll
    for (int i = 0; i < 3; ++i)
#pragma unroll
      for (int j = 0; j < 3; ++j)
        Amat[i][j] = Hm[0][i] * Hm[0][j] + Hm[1][i] * Hm[1][j] + Hm[2][i] * Hm[2][j];

    float Vm[3][3] = {{1.f, 0.f, 0.f}, {0.f, 1.f, 0.f}, {0.f, 0.f, 1.f}};

#define JROT(p, q, r)                                                          \
  {                                                                            \
    float apq = Amat[p][q];                                                    \
    if (fabsf(apq) > 1e-20f) {                                                 \
      float tau = (Amat[q][q] - Amat[p][p]) / (2.0f * apq);                    \
      float tt = copysignf(1.0f, tau) / (fabsf(tau) + sqrtf(1.0f + tau * tau));\
      float cc = 1.0f / sqrtf(1.0f + tt * tt);                                 \
      float ss = tt * cc;                                                      \
      float app = Amat[p][p], aqq = Amat[q][q];                                \
      float apr = Amat[p][r], aqr = Amat[q][r];                                \
      Amat[p][p] = app - tt * apq;                                             \
      Amat[q][q] = aqq + tt * apq;                                             \
      Amat[p][q] = 0.0f; Amat[q][p] = 0.0f;                                    \
      float npr = cc * apr - ss * aqr;                                         \
      float nqr = ss * apr + cc * aqr;                                         \
      Amat[p][r] = npr; Amat[r][p] = npr;                                      \
      Amat[q][r] = nqr; Amat[r][q] = nqr;                                      \
      _Pragma("unroll") for (int i_ = 0; i_ < 3; ++i_) {                       \
        float vp = Vm[i_][p], vq = Vm[i_][q];                                  \
        Vm[i_][p] = cc * vp - ss * vq;                                         \
        Vm[i_][q] = ss * vp + cc * vq;                                         \
      }                                                                        \
    }                                                                          \
  }
#pragma unroll
    for (int swp = 0; swp < 6; ++swp) { JROT(0, 1, 2) JROT(0, 2, 1) JROT(1, 2, 0) }
#undef JROT

    float eig[3] = {Amat[0][0], Amat[1][1], Amat[2][2]};
#define CSWAP(ia, ib)                                                          \
    if (eig[ia] < eig[ib]) {                                                   \
      float e_ = eig[ia]; eig[ia] = eig[ib]; eig[ib] = e_;                     \
      _Pragma("unroll") for (int i_ = 0; i_ < 3; ++i_) {                       \
        float t_ = Vm[i_][ia]; Vm[i_][ia] = Vm[i_][ib]; Vm[i_][ib] = t_;       \
      }                                                                        \
    }
    CSWAP(0, 1) CSWAP(1, 2) CSWAP(0, 1)
#undef CSWAP

    float b1[3], b2[3];
#pragma unroll
    for (int i = 0; i < 3; ++i) {
      b1[i] = Hm[i][0] * Vm[0][0] + Hm[i][1] * Vm[1][0] + Hm[i][2] * Vm[2][0];
      b2[i] = Hm[i][0] * Vm[0][1] + Hm[i][1] * Vm[1][1] + Hm[i][2] * Vm[2][1];
    }
    float n1 = b1[0] * b1[0] + b1[1] * b1[1] + b1[2] * b1[2];
    float r1 = 1.0f / sqrtf(fmaxf(n1, 1e-30f));
    float u1[3] = {b1[0] * r1, b1[1] * r1, b1[2] * r1};
    float d12 = u1[0] * b2[0] + u1[1] * b2[1] + u1[2] * b2[2];
    float u2[3] = {b2[0] - d12 * u1[0], b2[1] - d12 * u1[1], b2[2] - d12 * u1[2]};
    float n2 = u2[0] * u2[0] + u2[1] * u2[1] + u2[2] * u2[2];
    float r2 = 1.0f / sqrtf(fmaxf(n2, 1e-30f));
    u2[0] *= r2; u2[1] *= r2; u2[2] *= r2;
    float u3[3] = {u1[1] * u2[2] - u1[2] * u2[1],
                   u1[2] * u2[0] - u1[0] * u2[2],
                   u1[0] * u2[1] - u1[1] * u2[0]};
    float detV = Vm[0][0] * (Vm[1][1] * Vm[2][2] - Vm[1][2] * Vm[2][1]) -
                 Vm[0][1] * (Vm[1][0] * Vm[2][2] - Vm[1][2] * Vm[2][0]) +
                 Vm[0][2] * (Vm[1][0] * Vm[2][1] - Vm[1][1] * Vm[2][0]);
    float d3 = copysignf(1.0f, detV);

    float R[3][3];
#pragma unroll
    for (int i = 0; i < 3; ++i)
#pragma unroll
      for (int j = 0; j < 3; ++j)
        R[i][j] = Vm[i][0] * u1[j] + Vm[i][1] * u2[j] + d3 * Vm[i][2] * u3[j];

    float tv[3];
#pragma unroll
    for (int i = 0; i < 3; ++i)
      tv[i] = mut[i] - (R[i][0] * mus[0] + R[i][1] * mus[1] + R[i][2] * mus[2]);

    float* orow = outR + (size_t)b * 9;
#pragma unroll
    for (int i = 0; i < 3; ++i)
#pragma unroll
      for (int j = 0; j < 3; ++j) orow[i * 3 + j] = R[i][j];
    float* otr = outT + (size_t)b * 3;
    otr[0] = tv[0]; otr[1] = tv[1]; otr[2] = tv[2];
  }
}

extern "C" void kernel_launch(void* const* d_in, const int* in_sizes, int n_in,
                              void* d_out, int out_size, void* d_ws, size_t ws_size,
                              hipStream_t stream) {
  const float* src = (const float*)d_in[0];
  const float* tgt = (const float*)d_in[1];
  const float* wts = (const float*)d_in[2];
  float* out = (float*)d_out;
  const int B = out_size / 12;           // 9 (R) + 3 (t) per batch
  const int N = in_sizes[2] / B;         // weights are (B, N)
  float* outR = out;
  float* outT = out + (size_t)B * 9;
  procrustes_fused<<<B, 256, 0, stream>>>(src, tgt, wts, outR, outT, N);
}